// ContextBlock_7378753814694
// MI455X (gfx1250) — compile-verified
//
#include <hip/hip_runtime.h>
#include <hip/hip_bf16.h>
#include <math.h>

// ---------------- problem constants ----------------
#define N_  64
#define C_  256
#define P_  64
#define H_  48
#define W_  48
#define HW_ 2304
#define D_  4000
#define EPS_IN 1e-5f
#define EPS_BN 1e-5f

typedef float v2f __attribute__((ext_vector_type(2)));
typedef float v8f __attribute__((ext_vector_type(8)));

// D = A(16x4) * B(4x16) + C, fp32 WMMA, wave32.
// Lane layout (ISA 7.12.2, 32-bit operands): lid=lane&15, half=lane>>4.
//   A: v0 = A[lid][2*half], v1 = A[lid][2*half+1]
//   B: v0 = B[2*half][lid], v1 = B[2*half+1][lid]
//   C/D: v[j] = C[j + 8*half][lid]
__device__ __forceinline__ v8f wmma4(v2f a, v2f b, v8f c) {
    return __builtin_amdgcn_wmma_f32_16x16x4_f32(false, a, false, b, (short)0, c,
                                                 false, false);
}

__device__ __forceinline__ float sigmoidf_(float x) {
    return 1.0f / (1.0f + expf(-x));
}

// ---- CDNA5 async global->LDS DMA (ASYNCcnt-tracked), inline asm ----
// Flat pointers to __shared__ have the wave-relative LDS offset in the low
// 32 bits (ISA 10.2 aperture mapping), so truncation gives the VDST operand.
__device__ __forceinline__ unsigned lds_addr32(const void* p) {
    return (unsigned)(unsigned long long)p;
}
__device__ __forceinline__ void async_ld_b128(void* lds, const void* g) {
    asm volatile("global_load_async_to_lds_b128 %0, %1, off"
                 :: "v"(lds_addr32(lds)), "v"((unsigned long long)g)
                 : "memory");
}
#define WAIT_ASYNC(n) asm volatile("s_wait_asynccnt %0" :: "i"(n) : "memory")

// ---------------- K1: spectral norms via power iteration on G = W W^T ----------
__global__ __launch_bounds__(256) void k_sigma(const float* __restrict__ Wx,
                                               const float* __restrict__ Wy,
                                               float* __restrict__ invsig) {
    const float* Wm = blockIdx.x ? Wy : Wx;   // (64 x 256) row-major
    __shared__ float G[64][65];
    __shared__ float v[64], u[64];
    __shared__ float lamS;
    int t = threadIdx.x;
    for (int i = 0; i < 16; ++i) {
        int idx = t + 256 * i;
        int r = idx >> 6, c = idx & 63;
        float s = 0.f;
        for (int k = 0; k < C_; ++k) s += Wm[r * C_ + k] * Wm[c * C_ + k];
        G[r][c] = s;
    }
    if (t < 64) v[t] = 0.125f;
    __syncthreads();
    for (int it = 0; it < 300; ++it) {
        if (t < 64) {
            float s = 0.f;
            for (int j = 0; j < 64; ++j) s += G[t][j] * v[j];
            u[t] = s;
        }
        __syncthreads();
        if (t == 0) {
            float s = 0.f;
            for (int j = 0; j < 64; ++j) s += u[j] * u[j];
            lamS = sqrtf(s);           // -> lambda_max(G) = sigma_max(W)^2
        }
        __syncthreads();
        if (t < 64) v[t] = u[t] / lamS;
        __syncthreads();
    }
    if (t == 0) invsig[blockIdx.x] = 1.0f / sqrtf(lamS);
}

// ---------------- K2: context GEMM  ctx[n] = (W@x[n]) * (1/sigma) --------------
// A(64x256) async-DMA'd raw (scale folded into the store), B strips 256x32,
// both double-buffered in LDS via GLOBAL_LOAD_ASYNC_TO_LDS_B128.
__global__ __launch_bounds__(256) void k_ctx(const float* __restrict__ x,
                                             const float* __restrict__ x2,
                                             const float* __restrict__ Wx,
                                             const float* __restrict__ Wy,
                                             const float* __restrict__ invsig,
                                             float* __restrict__ ctxA,
                                             float* __restrict__ ctxV) {
    int strip = blockIdx.x;   // 0..71 (32-wide column strips)
    int n     = blockIdx.y;   // 0..63
    int which = blockIdx.z;   // 0: A, 1: V
    const float* X  = which ? x2 : x;
    const float* Wm = which ? Wy : Wx;
    float isg       = invsig[which];
    float* out      = which ? ctxV : ctxA;

    __shared__ float As[2][64][68];   // stride 68: 16B-aligned rows, 4r-bank spread
    __shared__ float Bs[2][64][40];   // stride 40: half-wave bank shift 16

    int t = threadIdx.x;
    int lane = t & 31, wv = t >> 5;
    int lid = lane & 15, half = lane >> 4;
    int rt = wv >> 1, ct = wv & 1;          // 4 row tiles x 2 col tiles
    int colbase = strip * 32;
    const float* Xn = X + (size_t)n * C_ * HW_;

    auto issueA = [&](int k0, int b) {
        #pragma unroll
        for (int i = 0; i < 4; ++i) {           // 64x64 floats, 4x b128/thread
            int idx = t + 256 * i;
            int r = idx >> 4, q = idx & 15;
            async_ld_b128(&As[b][r][q * 4], Wm + r * C_ + k0 + q * 4);
        }
    };
    auto issueB = [&](int k0, int b) {
        #pragma unroll
        for (int i = 0; i < 2; ++i) {           // 64x32 floats, 2x b128/thread
            int idx = t + 256 * i;
            int r = idx >> 3, q = idx & 7;
            async_ld_b128(&Bs[b][r][q * 4],
                          Xn + (size_t)(k0 + r) * HW_ + colbase + q * 4);
        }
    };

    v8f acc = {0.f, 0.f, 0.f, 0.f, 0.f, 0.f, 0.f, 0.f};
    issueA(0, 0); issueB(0, 0);                 // 6 ops in flight
    for (int kc = 0; kc < 4; ++kc) {
        int k0 = kc * 64, b = kc & 1;
        if (kc < 3) {
            issueA(k0 + 64, 1 - b); issueB(k0 + 64, 1 - b);
            WAIT_ASYNC(6);                      // oldest 6 (= current buffer) done
        } else {
            WAIT_ASYNC(0);
        }
        __syncthreads();
        #pragma unroll
        for (int kk = 0; kk < 16; ++kk) {
            int kb = kk * 4 + 2 * half;
            v2f a = { As[b][rt * 16 + lid][kb], As[b][rt * 16 + lid][kb + 1] };
            v2f bv = { Bs[b][kb][ct * 16 + lid], Bs[b][kb + 1][ct * 16 + lid] };
            acc = wmma4(a, bv, acc);
        }
        __syncthreads();
    }
    float* On = out + (size_t)n * P_ * HW_;
    #pragma unroll
    for (int j = 0; j < 8; ++j) {
        int row = rt * 16 + j + 8 * half;
        On[(size_t)row * HW_ + colbase + ct * 16 + lid] = acc[j] * isg;
    }
}

// ---------------- K3: 3x3 adaptive average pool (16x16 bins) --------------------
__global__ __launch_bounds__(288) void k_gap(const float* __restrict__ ctxA,
                                             const float* __restrict__ ctxV,
                                             float* __restrict__ gA,
                                             float* __restrict__ gV) {
    int np = blockIdx.x;            // n*64 + p
    int which = blockIdx.y;
    int n = np >> 6, p = np & 63;
    const float* src = (which ? ctxV : ctxA) + ((size_t)n * P_ + p) * HW_;
    float* dst       = (which ? gV : gA) + ((size_t)n * P_ + p) * 9;
    int t = threadIdx.x, wv = t >> 5, lane = t & 31;  // 9 waves, one bin each
    int bi = wv / 3, bj = wv % 3;
    float s = 0.f;
    #pragma unroll
    for (int i = 0; i < 8; ++i) {
        int idx = i * 32 + lane;                // two contiguous 16-float rows/load
        int rr = idx >> 4, cc = idx & 15;
        s += src[(bi * 16 + rr) * W_ + bj * 16 + cc];
    }
    #pragma unroll
    for (int off = 16; off > 0; off >>= 1) s += __shfl_down(s, off, 32);
    if (lane == 0) dst[wv] = s * (1.0f / 256.0f);
}

// ---------------- K4: normalize gA/gV over the batch dim (in place) -------------
__global__ __launch_bounds__(576) void k_normg(float* __restrict__ gA,
                                               float* __restrict__ gV) {
    float* g = blockIdx.x ? gV : gA;
    int t = threadIdx.x;                        // p*9+l index, 0..575
    float s = 0.f;
    for (int n = 0; n < N_; ++n) { float v = g[n * 576 + t]; s += v * v; }
    float inv = 1.0f / sqrtf(s);
    for (int n = 0; n < N_; ++n) g[n * 576 + t] *= inv;
}

// ---------------- K5: sparse compact bilinear pooling (replaces FFT) ------------
__global__ __launch_bounds__(256) void k_cbp(const float* __restrict__ gA,
                                             const float* __restrict__ gV,
                                             const int* __restrict__ h1,
                                             const int* __restrict__ h2,
                                             const int* __restrict__ s1,
                                             const int* __restrict__ s2,
                                             float* __restrict__ att) {
    int n = blockIdx.x;
    __shared__ float attL[D_];
    __shared__ float av[9][64], bv[9][64];
    __shared__ int hs1[64], hs2[64];
    int t = threadIdx.x;
    for (int i = t; i < D_; i += 256) attL[i] = 0.f;
    for (int i = t; i < 576; i += 256) {
        int p = i / 9, l = i % 9;
        av[l][p] = gA[n * 576 + p * 9 + l] * (float)(2 * s1[p] - 1);
        bv[l][p] = gV[n * 576 + p * 9 + l] * (float)(2 * s2[p] - 1);
    }
    if (t < 64) { hs1[t] = h1[t]; hs2[t] = h2[t]; }
    __syncthreads();
    for (int l = 0; l < 9; ++l) {
        #pragma unroll 4
        for (int i = 0; i < 16; ++i) {
            int idx = t + 256 * i;
            int p = idx >> 6, q = idx & 63;
            int bin = hs1[p] + hs2[q];
            if (bin >= D_) bin -= D_;
            atomicAdd(&attL[bin], av[l][p] * bv[l][q]);  // ds_add_f32
        }
    }
    __syncthreads();
    for (int i = t; i < D_; i += 256) att[(size_t)n * D_ + i] = attL[i];
}

// ---------------- K6: fc1 via WMMA, K split over 8 waves, 32 blocks -------------
__global__ __launch_bounds__(256) void k_fc1(const float* __restrict__ att,
                                             const float* __restrict__ fc1_w,
                                             const float* __restrict__ fc1_b,
                                             float* __restrict__ z) {
    int tile = blockIdx.x;          // 0..31: 4 M-tiles x 8 N-tiles
    int rt = tile & 3, ct = tile >> 2;
    int t = threadIdx.x, lane = t & 31, wv = t >> 5;
    int lid = lane & 15, half = lane >> 4;
    __shared__ float part[8][260];
    v8f acc = {0.f, 0.f, 0.f, 0.f, 0.f, 0.f, 0.f, 0.f};
    const float* arow = att + (size_t)(rt * 16 + lid) * D_ + wv * 500;
    const float* brow = fc1_w + (size_t)(ct * 16 + lid) * D_ + wv * 500;
    for (int k = 0; k < 500; k += 4) {          // 125 WMMA per wave
        v2f a = { arow[k + 2 * half], arow[k + 2 * half + 1] };
        v2f b = { brow[k + 2 * half], brow[k + 2 * half + 1] };
        acc = wmma4(a, b, acc);
    }
    #pragma unroll
    for (int j = 0; j < 8; ++j) part[j][wv * 32 + lane] = acc[j];
    __syncthreads();
    if (wv == 0) {                               // deterministic partial reduce
        #pragma unroll
        for (int j = 0; j < 8; ++j) {
            float s = 0.f;
            #pragma unroll
            for (int w8 = 0; w8 < 8; ++w8) s += part[j][w8 * 32 + lane];
            int row = rt * 16 + j + 8 * half;
            int col = ct * 16 + lid;
            z[row * 128 + col] = s + fc1_b[col];
        }
    }
}

// ---------------- K7: BatchNorm(batch stats) + LeakyReLU + fc2 + tanh -----------
__global__ __launch_bounds__(128) void k_bn_e(const float* __restrict__ z,
                                              const float* __restrict__ bn_g,
                                              const float* __restrict__ bn_b,
                                              const float* __restrict__ fc2_w,
                                              const float* __restrict__ fc2_b,
                                              float* __restrict__ ebuf,
                                              float* __restrict__ out_e) {
    __shared__ float mS[128], isS[128], gS[128], bS[128];
    int t = threadIdx.x;
    float s = 0.f;
    for (int n = 0; n < N_; ++n) s += z[n * 128 + t];
    float m = s * (1.0f / N_);
    float vq = 0.f;
    for (int n = 0; n < N_; ++n) { float d = z[n * 128 + t] - m; vq += d * d; }
    vq *= (1.0f / N_);
    mS[t] = m;
    isS[t] = 1.0f / sqrtf(vq + EPS_BN);
    gS[t] = bn_g[t];
    bS[t] = bn_b[t];
    __syncthreads();
    if (t < N_) {
        float acc = 0.f;
        for (int c = 0; c < 128; ++c) {
            float zb = (z[t * 128 + c] - mS[c]) * isS[c] * gS[c] + bS[c];
            zb = zb >= 0.f ? zb : 0.2f * zb;
            acc += zb * fc2_w[c];
        }
        float ev = tanhf(acc + fc2_b[0]);
        ebuf[t] = ev;
        out_e[t] = ev;
    }
}

// ---------------- K8: cross = softmax(ctx @ ctx2^T); M = W_add[:,1:] @ cross ----
__global__ __launch_bounds__(256) void k_cross(const float* __restrict__ gA,
                                               const float* __restrict__ gV,
                                               const float* __restrict__ W_add,
                                               float* __restrict__ Mbuf) {
    int n = blockIdx.x;
    __shared__ float c1[64][10], c2[64][10];
    __shared__ float S[64][65];
    int t = threadIdx.x;
    for (int i = t; i < 576; i += 256) {
        int p = i / 9, l = i % 9;
        c1[p][l] = gA[n * 576 + p * 9 + l];
        c2[p][l] = gV[n * 576 + p * 9 + l];
    }
    __syncthreads();
    for (int i = 0; i < 16; ++i) {
        int idx = t + 256 * i;
        int p = idx >> 6, q = idx & 63;
        float s = 0.f;
        #pragma unroll
        for (int l = 0; l < 9; ++l) s += c1[p][l] * c2[q][l];
        S[p][q] = s;
    }
    __syncthreads();
    if (t < 64) {                               // row softmax
        float mx = -1e30f;
        for (int q = 0; q < 64; ++q) mx = fmaxf(mx, S[t][q]);
        float sum = 0.f;
        for (int q = 0; q < 64; ++q) { float ex = expf(S[t][q] - mx); S[t][q] = ex; sum += ex; }
        float inv = 1.0f / sum;
        for (int q = 0; q < 64; ++q) S[t][q] *= inv;
    }
    __syncthreads();
    // M(256x64) = W_add[:,1:](256x64) @ S(64x64)  via WMMA
    int lane = t & 31, wv = t >> 5;
    int lid = lane & 15, half = lane >> 4;
    float* Mn = Mbuf + (size_t)n * C_ * P_;
    for (int tile = wv; tile < 64; tile += 8) {   // 16 M-tiles x 4 N-tiles
        int rt = tile >> 2, ct = tile & 3;
        v8f acc = {0.f, 0.f, 0.f, 0.f, 0.f, 0.f, 0.f, 0.f};
        const float* arow = W_add + (size_t)(rt * 16 + lid) * (P_ + 1) + 1;
        #pragma unroll
        for (int k = 0; k < 64; k += 4) {
            v2f a = { arow[k + 2 * half], arow[k + 2 * half + 1] };
            v2f b = { S[k + 2 * half][ct * 16 + lid], S[k + 2 * half + 1][ct * 16 + lid] };
            acc = wmma4(a, b, acc);
        }
        #pragma unroll
        for (int j = 0; j < 8; ++j)
            Mn[(rt * 16 + j + 8 * half) * P_ + ct * 16 + lid] = acc[j];
    }
}

// ---------------- K9: pass 1 — Y = sigmoid(M@ctx + bias); per-(n,c) stats -------
// B strips double-buffered with async DMA; A DMA'd once; deterministic row stats.
__global__ __launch_bounds__(256) void k_pass1(const float* __restrict__ ctxA,
                                               const float* __restrict__ ctxV,
                                               const float* __restrict__ Mbuf,
                                               const float* __restrict__ W_add,
                                               const float* __restrict__ ebuf,
                                               float* __restrict__ st_sum,
                                               float* __restrict__ st_sq,
                                               float* __restrict__ st_mx) {
    int rg    = blockIdx.x;   // 0..3   (64-row group of C)
    int n     = blockIdx.y;   // 0..63
    int which = blockIdx.z;   // 0..1
    const float* B  = (which ? ctxV : ctxA) + (size_t)n * P_ * HW_;
    const float* Mn = Mbuf + (size_t)n * C_ * P_ + (size_t)rg * 64 * P_;

    __shared__ float As[64][68];
    __shared__ float Bs[2][64][40];
    __shared__ float Ys[64][33];
    __shared__ float bias[64];

    int t = threadIdx.x;
    auto issueB = [&](int strip, int b) {
        #pragma unroll
        for (int i = 0; i < 2; ++i) {
            int idx = t + 256 * i;
            int r = idx >> 3, q = idx & 7;
            async_ld_b128(&Bs[b][r][q * 4],
                          B + (size_t)r * HW_ + strip * 32 + q * 4);
        }
    };
    #pragma unroll
    for (int i = 0; i < 4; ++i) {               // A: 64x64 floats via DMA
        int idx = t + 256 * i;
        int r = idx >> 4, q = idx & 15;
        async_ld_b128(&As[r][q * 4], Mn + r * P_ + q * 4);
    }
    issueB(0, 0);                               // 6 ops in flight
    if (t < 64) bias[t] = W_add[(rg * 64 + t) * (P_ + 1)] * ebuf[n];

    int lane = t & 31, wv = t >> 5, lid = lane & 15, half = lane >> 4;
    int rt = wv >> 1, ct = wv & 1;
    float ssum = 0.f, ssq = 0.f, smx = 0.f;     // sigmoid>0 so max init 0 is safe

    for (int strip = 0; strip < 72; ++strip) {
        int b = strip & 1;
        if (strip < 71) {
            issueB(strip + 1, 1 - b);
            WAIT_ASYNC(2);                      // current strip (oldest) landed
        } else {
            WAIT_ASYNC(0);
        }
        __syncthreads();
        v8f acc = {0.f, 0.f, 0.f, 0.f, 0.f, 0.f, 0.f, 0.f};
        #pragma unroll
        for (int kk = 0; kk < 16; ++kk) {
            int kb = kk * 4 + 2 * half;
            v2f a = { As[rt * 16 + lid][kb], As[rt * 16 + lid][kb + 1] };
            v2f bv = { Bs[b][kb][ct * 16 + lid], Bs[b][kb + 1][ct * 16 + lid] };
            acc = wmma4(a, bv, acc);
        }
        #pragma unroll
        for (int j = 0; j < 8; ++j) {
            int row = rt * 16 + j + 8 * half;
            Ys[row][ct * 16 + lid] = sigmoidf_(acc[j] + bias[row]);
        }
        __syncthreads();
        if (t < 64) {                           // deterministic per-row reduction
            #pragma unroll 8
            for (int c = 0; c < 32; ++c) {
                float y = Ys[t][c];
                ssum += y; ssq += y * y; smx = fmaxf(smx, y);
            }
        }
    }
    if (t < 64) {
        size_t o = ((size_t)(which * N_ + n)) * C_ + rg * 64 + t;
        st_sum[o] = ssum; st_sq[o] = ssq; st_mx[o] = smx;
    }
}

// ---------------- K10: channel-attention mask + instnorm coefficients -----------
__global__ __launch_bounds__(256) void k_mask(const float* __restrict__ st_sum,
                                              const float* __restrict__ st_sq,
                                              const float* __restrict__ st_mx,
                                              const float* __restrict__ Wm1,
                                              const float* __restrict__ Wm2,
                                              float* __restrict__ meanb,
                                              float* __restrict__ invstdb,
                                              float* __restrict__ maskb) {
    int b = blockIdx.x;                 // 0..127
    int which = b & 1, n = b >> 1;
    __shared__ float avgS[256], mxS[256], hA[64], hM[64];
    int t = threadIdx.x;
    size_t base = ((size_t)(which * N_ + n)) * C_;
    float mean = st_sum[base + t] * (1.0f / HW_);
    float var  = st_sq[base + t] * (1.0f / HW_) - mean * mean;
    if (var < 0.f) var = 0.f;
    meanb[base + t]   = mean;
    invstdb[base + t] = 1.0f / sqrtf(var + EPS_IN);
    avgS[t] = mean;
    mxS[t]  = st_mx[base + t];
    __syncthreads();
    if (t < 64) {
        float ha = 0.f, hm = 0.f;
        for (int c = 0; c < C_; ++c) {
            float w = Wm1[t * C_ + c];
            ha += w * avgS[c]; hm += w * mxS[c];
        }
        hA[t] = fmaxf(ha, 0.f); hM[t] = fmaxf(hm, 0.f);
    }
    __syncthreads();
    float oa = 0.f, om = 0.f;
    for (int h = 0; h < 64; ++h) {
        float w = Wm2[t * 64 + h];
        oa += w * hA[h]; om += w * hM[h];
    }
    maskb[base + t] = sigmoidf_(oa) + sigmoidf_(om);
}

// ---------------- K11: pass 2 — recompute Y, fuse instnorm + mask, write out ----
__global__ __launch_bounds__(256) void k_pass2(const float* __restrict__ x,
                                               const float* __restrict__ x2,
                                               const float* __restrict__ ctxA,
                                               const float* __restrict__ ctxV,
                                               const float* __restrict__ Mbuf,
                                               const float* __restrict__ W_add,
                                               const float* __restrict__ ebuf,
                                               const float* __restrict__ meanb,
                                               const float* __restrict__ invstdb,
                                               const float* __restrict__ maskb,
                                               float* __restrict__ out) {
    int strip = blockIdx.x;  // 0..71
    int rg    = blockIdx.y;  // 0..3
    int nz    = blockIdx.z;  // 0..127
    int n = nz >> 1, which = nz & 1;
    const float* B  = (which ? ctxV : ctxA) + (size_t)n * P_ * HW_;
    const float* X  = which ? x2 : x;
    const float* Mn = Mbuf + (size_t)n * C_ * P_ + (size_t)rg * 64 * P_;
    float* O = out + (size_t)which * N_ * C_ * HW_;

    __shared__ float As[64][68];
    __shared__ float Bs[64][40];
    __shared__ float bias[64], mS[64], isS[64], kS[64];

    int t = threadIdx.x;
    #pragma unroll
    for (int i = 0; i < 4; ++i) {               // A via async DMA
        int idx = t + 256 * i;
        int r = idx >> 4, q = idx & 15;
        async_ld_b128(&As[r][q * 4], Mn + r * P_ + q * 4);
    }
    #pragma unroll
    for (int i = 0; i < 2; ++i) {               // B strip via async DMA
        int idx = t + 256 * i;
        int r = idx >> 3, q = idx & 7;
        async_ld_b128(&Bs[r][q * 4], B + (size_t)r * HW_ + strip * 32 + q * 4);
    }
    if (t < 64) {
        int c = rg * 64 + t;
        size_t o = ((size_t)(which * N_ + n)) * C_ + c;
        bias[t] = W_add[c * (P_ + 1)] * ebuf[n];
        mS[t] = meanb[o]; isS[t] = invstdb[o]; kS[t] = maskb[o];
    }
    WAIT_ASYNC(0);
    __syncthreads();

    int lane = t & 31, wv = t >> 5, lid = lane & 15, half = lane >> 4;
    int rt = wv >> 1, ct = wv & 1;
    v8f acc = {0.f, 0.f, 0.f, 0.f, 0.f, 0.f, 0.f, 0.f};
    #pragma unroll
    for (int kk = 0; kk < 16; ++kk) {
        int kb = kk * 4 + 2 * half;
        v2f a = { As[rt * 16 + lid][kb], As[rt * 16 + lid][kb + 1] };
        v2f b = { Bs[kb][ct * 16 + lid], Bs[kb + 1][ct * 16 + lid] };
        acc = wmma4(a, b, acc);
    }
    #pragma unroll
    for (int j = 0; j < 8; ++j) {
        int row = rt * 16 + j + 8 * half;
        int c   = rg * 64 + row;
        int col = strip * 32 + ct * 16 + lid;
        float y  = sigmoidf_(acc[j] + bias[row]);
        float mk = kS[row];
        size_t off = ((size_t)n * C_ + c) * HW_ + col;
        float xv = __builtin_nontemporal_load(&X[off]);
        __builtin_nontemporal_store(xv * mk + (y - mS[row]) * isS[row] * (1.0f - mk),
                                    &O[off]);
    }
}

// ---------------- launch ----------------
extern "C" void kernel_launch(void* const* d_in, const int* in_sizes, int n_in,
                              void* d_out, int out_size, void* d_ws, size_t ws_size,
                              hipStream_t stream) {
    const float* x       = (const float*)d_in[0];
    const float* x2      = (const float*)d_in[1];
    const float* W_x     = (const float*)d_in[2];
    const float* W_y     = (const float*)d_in[3];
    const float* W_mam1  = (const float*)d_in[4];
    const float* W_mam2  = (const float*)d_in[5];
    const float* W_add   = (const float*)d_in[6];
    const float* fc1_w   = (const float*)d_in[7];
    const float* fc1_b   = (const float*)d_in[8];
    const float* bn_g    = (const float*)d_in[9];
    const float* bn_b    = (const float*)d_in[10];
    const float* fc2_w   = (const float*)d_in[11];
    const float* fc2_b   = (const float*)d_in[12];
    const int*   h1      = (const int*)d_in[13];
    const int*   h2      = (const int*)d_in[14];
    const int*   s1      = (const int*)d_in[15];
    const int*   s2      = (const int*)d_in[16];
    float* out = (float*)d_out;

    const size_t NCHW = (size_t)N_ * C_ * HW_;

    float* w = (float*)d_ws;
    float* invsig = w;  w += 16;
    float* ctxA   = w;  w += (size_t)N_ * P_ * HW_;
    float* ctxV   = w;  w += (size_t)N_ * P_ * HW_;
    float* gA     = w;  w += (size_t)N_ * P_ * 9;
    float* gV     = w;  w += (size_t)N_ * P_ * 9;
    float* att    = w;  w += (size_t)N_ * D_;
    float* z      = w;  w += (size_t)N_ * 128;
    float* ebuf   = w;  w += 64;
    float* Mbuf   = w;  w += (size_t)N_ * C_ * P_;
    float* st_sum = w;  w += (size_t)2 * N_ * C_;
    float* st_sq  = w;  w += (size_t)2 * N_ * C_;
    float* st_mx  = w;  w += (size_t)2 * N_ * C_;
    float* meanb  = w;  w += (size_t)2 * N_ * C_;
    float* invstdb= w;  w += (size_t)2 * N_ * C_;
    float* maskb  = w;  w += (size_t)2 * N_ * C_;

    k_sigma<<<2, 256, 0, stream>>>(W_x, W_y, invsig);
    k_ctx<<<dim3(72, N_, 2), 256, 0, stream>>>(x, x2, W_x, W_y, invsig, ctxA, ctxV);
    k_gap<<<dim3(N_ * P_, 2), 288, 0, stream>>>(ctxA, ctxV, gA, gV);
    k_normg<<<2, 576, 0, stream>>>(gA, gV);
    k_cbp<<<N_, 256, 0, stream>>>(gA, gV, h1, h2, s1, s2, att);
    k_fc1<<<32, 256, 0, stream>>>(att, fc1_w, fc1_b, z);
    k_bn_e<<<1, 128, 0, stream>>>(z, bn_g, bn_b, fc2_w, fc2_b, ebuf, out + 2 * NCHW);
    k_cross<<<N_, 256, 0, stream>>>(gA, gV, W_add, Mbuf);
    k_pass1<<<dim3(4, N_, 2), 256, 0, stream>>>(ctxA, ctxV, Mbuf, W_add, ebuf,
                                                st_sum, st_sq, st_mx);
    k_mask<<<2 * N_, 256, 0, stream>>>(st_sum, st_sq, st_mx, W_mam1, W_mam2,
                                       meanb, invstdb, maskb);
    k_pass2<<<dim3(72, 4, 2 * N_), 256, 0, stream>>>(x, x2, ctxA, ctxV, Mbuf, W_add,
                                                     ebuf, meanb, invstdb, maskb, out);
    (void)in_sizes; (void)n_in; (void)out_size; (void)ws_size;
}